// BioClassifier_58162447122741
// MI455X (gfx1250) — compile-verified
//
#include <hip/hip_runtime.h>
#include <math.h>

typedef __attribute__((ext_vector_type(2))) float v2f;
typedef __attribute__((ext_vector_type(8))) float v8f;

#define B_    16
#define D_    1024
#define H_    4096
#define NSEL  17          // K+1
#define DELTA 0.4f

// ---------------------------------------------------------------------------
// Kernel 1: top-17 indices of each x row (one wave32 per sample).
// Each lane caches 32 strided elements; 17 rounds of wave argmax with masking.
// Tie-break: larger value, then lower index (matches lax.top_k).
// ---------------------------------------------------------------------------
__global__ __launch_bounds__(32) void bio_topk(const float* __restrict__ x,
                                               int* __restrict__ idx_out) {
    const int b = blockIdx.x;
    const int lane = threadIdx.x;
    float v[32];
#pragma unroll
    for (int j = 0; j < 32; ++j) v[j] = x[b * D_ + j * 32 + lane];

    for (int it = 0; it < NSEL; ++it) {
        float bestv = -__builtin_inff();
        int   besti = 0x7fffffff;
#pragma unroll
        for (int j = 0; j < 32; ++j) {
            const int id = j * 32 + lane;
            if (v[j] > bestv || (v[j] == bestv && id < besti)) { bestv = v[j]; besti = id; }
        }
        for (int off = 16; off >= 1; off >>= 1) {
            const float ov = __shfl_xor(bestv, off, 32);
            const int   oi = __shfl_xor(besti, off, 32);
            if (ov > bestv || (ov == bestv && oi < besti)) { bestv = ov; besti = oi; }
        }
        if ((besti & 31) == lane) v[besti >> 5] = -__builtin_inff();
        if (lane == 0) idx_out[b * 32 + it] = besti;
    }
}

// ---------------------------------------------------------------------------
// Kernel 2: zero-fill the 256 MB output (the bandwidth-dominant phase).
// Grid-stride B128 stores.
// ---------------------------------------------------------------------------
__global__ __launch_bounds__(256) void bio_zero(float4* __restrict__ out, long n4) {
    long i = (long)blockIdx.x * 256 + threadIdx.x;
    const long stride = (long)gridDim.x * 256;
    const float4 z = make_float4(0.f, 0.f, 0.f, 0.f);
    for (; i < n4; i += stride) out[i] = z;
}

// ---------------------------------------------------------------------------
// Kernel 3: p_dot[b,h] = sum_d sign(W)|W|^2 * x  for h in [0,1024) via
// v_wmma_f32_16x16x4_f32. One wave per 16-wide h tile; K loop of 256 WMMAs.
// A tile (16x4 of x):  lanes 0-15 hold K=k,k+1; lanes 16-31 hold K=k+2,k+3.
// B tile (4x16 of sW^T): same half-wave K split, N striped across lanes.
// C/D: VGPR v -> batch row v (+8 on high lanes), N = lane&15.
// ---------------------------------------------------------------------------
__global__ __launch_bounds__(32) void bio_pdot_wmma(const float* __restrict__ x,
                                                    const float* __restrict__ W,
                                                    float* __restrict__ pdot) {
    const int lane = threadIdx.x;
    const int half = lane >> 4;      // 0 or 1
    const int m    = lane & 15;
    const int h    = blockIdx.x * 16 + m;

    v8f acc = {0.f, 0.f, 0.f, 0.f, 0.f, 0.f, 0.f, 0.f};
    const float* __restrict__ xrow = x + m * D_;       // batch row m (A matrix)
    const float* __restrict__ wrow = W + (size_t)h * D_;  // hidden row h (B matrix)

    for (int k = 0; k < D_; k += 4) {
        const int kk = k + 2 * half;
        v2f a, bm;
        a.x = xrow[kk];
        a.y = xrow[kk + 1];
        const float w0 = wrow[kk];
        const float w1 = wrow[kk + 1];
        bm.x = w0 * fabsf(w0);       // sign(W)|W|^(p-1), p=3
        bm.y = w1 * fabsf(w1);
        acc = __builtin_amdgcn_wmma_f32_16x16x4_f32(
            /*neg_a=*/false, a, /*neg_b=*/false, bm,
            /*c_mod=*/(short)0, acc, /*reuse_a=*/false, /*reuse_b=*/false);
    }
#pragma unroll
    for (int v = 0; v < 8; ++v) {
        const int brow = v + 8 * half;
        pdot[brow * D_ + h] = acc[v];
    }
}

// ---------------------------------------------------------------------------
// Kernel 4: per-sample max of g*bracket over the 17 active rows (zeros of the
// dense tensor are folded in via the 0.0f reduction identity).
// ---------------------------------------------------------------------------
__global__ __launch_bounds__(256) void bio_maxred(const float* __restrict__ x,
                                                  const float* __restrict__ W,
                                                  const float* __restrict__ pdot,
                                                  const int* __restrict__ idx,
                                                  float* __restrict__ maxima) {
    __shared__ float sm[256];
    const int b = blockIdx.x;
    const int tid = threadIdx.x;
    float local = 0.0f;                       // dense tensor is mostly zero
    for (int j = 0; j < NSEL; ++j) {
        const int   h  = idx[b * 32 + j];
        const float g  = (j == 0) ? 1.0f : -DELTA;
        const float pd = pdot[b * D_ + h];
        const float* __restrict__ wrow = W + (size_t)h * D_;
        const float* __restrict__ xrow = x + b * D_;
        for (int d = tid; d < D_; d += 256) {
            const float w = wrow[d];
            local = fmaxf(local, g * (fabsf(w) * xrow[d] - pd * w));
        }
    }
    sm[tid] = local;
    __syncthreads();
    for (int s = 128; s >= 1; s >>= 1) {
        if (tid < s) sm[tid] = fmaxf(sm[tid], sm[tid + s]);
        __syncthreads();
    }
    if (tid == 0) maxima[b] = sm[0];
}

// ---------------------------------------------------------------------------
// Kernel 5: write the 272 nonzero normalized rows into the zeroed output.
// ---------------------------------------------------------------------------
__global__ __launch_bounds__(256) void bio_rows(const float* __restrict__ x,
                                                const float* __restrict__ W,
                                                const float* __restrict__ pdot,
                                                const int* __restrict__ idx,
                                                const float* __restrict__ maxima,
                                                float* __restrict__ out) {
    const int blk = blockIdx.x;
    const int b = blk / NSEL;
    const int j = blk % NSEL;
    const int   h  = idx[b * 32 + j];
    const float g  = (j == 0) ? 1.0f : -DELTA;
    const float pd = pdot[b * D_ + h];
    const float mx = maxima[b];
    const int tid = threadIdx.x;
    const float* __restrict__ wrow = W + (size_t)h * D_;
    const float* __restrict__ xrow = x + b * D_;
    float* __restrict__ orow = out + (size_t)b * H_ * D_ + (size_t)h * D_;
    for (int d = tid; d < D_; d += 256) {
        const float w = wrow[d];
        orow[d] = (g * (fabsf(w) * xrow[d] - pd * w)) / mx;
    }
}

extern "C" void kernel_launch(void* const* d_in, const int* in_sizes, int n_in,
                              void* d_out, int out_size, void* d_ws, size_t ws_size,
                              hipStream_t stream) {
    const float* x = (const float*)d_in[0];   // [16, 1024]
    const float* W = (const float*)d_in[1];   // [4096, 1024]
    float* out = (float*)d_out;               // [16, 4096, 1024]

    char* ws = (char*)d_ws;
    float* pdot   = (float*)ws;                       // 16*1024 floats = 64 KB
    int*   idx    = (int*)(ws + 64 * 1024);           // 16*32 ints    =  2 KB
    float* maxima = (float*)(ws + 64 * 1024 + 2048);  // 16 floats

    // 1) competitive top-17 ranking of each input row
    bio_topk<<<B_, 32, 0, stream>>>(x, idx);

    // 2) zero-fill full dense output (bandwidth floor: 256 MB store)
    const long n4 = (long)out_size / 4;               // float4 count
    bio_zero<<<4096, 256, 0, stream>>>((float4*)out, n4);

    // 3) Lebesgue overlap GEMM (only h<1024 can be gated on) via WMMA f32
    bio_pdot_wmma<<<D_ / 16, 32, 0, stream>>>(x, W, pdot);

    // 4) per-sample global max of dW
    bio_maxred<<<B_, 256, 0, stream>>>(x, W, pdot, idx, maxima);

    // 5) scatter the 272 normalized nonzero rows
    bio_rows<<<B_ * NSEL, 256, 0, stream>>>(x, W, pdot, idx, maxima, out);
}